// GraphLayer_10934986735970
// MI455X (gfx1250) — compile-verified
//
#include <hip/hip_runtime.h>
#include <hip/hip_bf16.h>
#include <math.h>

// Problem constants (match reference)
#define BATCH 8
#define CIN   128
#define NPTS  4096
#define KNN   32
#define COUT  256

typedef float v2f __attribute__((ext_vector_type(2)));
typedef float v8f __attribute__((ext_vector_type(8)));

// ---------------------------------------------------------------------------
// Kernel 1: transpose x (B, C, N) -> xt (B, N, C), LDS-tiled 32x32
// ---------------------------------------------------------------------------
__global__ __launch_bounds__(256) void transpose_kernel(
    const float* __restrict__ x, float* __restrict__ xt) {
  __shared__ float tile[32][33];
  const int n0 = blockIdx.x * 32;
  const int c0 = blockIdx.y * 32;
  const int b  = blockIdx.z;
  const float* xb  = x  + (size_t)b * CIN * NPTS;
  float*       xtb = xt + (size_t)b * NPTS * CIN;
  #pragma unroll
  for (int i = threadIdx.y; i < 32; i += 8)
    tile[i][threadIdx.x] = xb[(size_t)(c0 + i) * NPTS + (n0 + threadIdx.x)];
  __syncthreads();
  #pragma unroll
  for (int i = threadIdx.y; i < 32; i += 8)
    xtb[(size_t)(n0 + i) * CIN + (c0 + threadIdx.x)] = tile[threadIdx.x][i];
}

// ---------------------------------------------------------------------------
// Kernel 2: gather + max-pool, float4-vectorized.
// One wave per point: lane L owns channels 4L..4L+3; each neighbor row is one
// coalesced 512 B (32 x b128) fetch.  xt batch slice (2 MB) is L2-resident.
// ---------------------------------------------------------------------------
__global__ __launch_bounds__(128) void pool_kernel(
    const float* __restrict__ xt, const int* __restrict__ idx,
    float* __restrict__ pooled) {
  const int pt   = blockIdx.x * 4 + (threadIdx.x >> 5);  // point in batch
  const int b    = blockIdx.y;
  const int lane = threadIdx.x & 31;
  const int*    ip = idx + ((size_t)b * NPTS + pt) * KNN;
  const float4* xb = (const float4*)(xt + (size_t)b * NPTS * CIN);
  float4 m = make_float4(-INFINITY, -INFINITY, -INFINITY, -INFINITY);
  #pragma unroll 8
  for (int k = 0; k < KNN; ++k) {
    const int j = ip[k];                     // uniform across wave
    const float4 v = xb[(size_t)j * (CIN / 4) + lane];
    m.x = fmaxf(m.x, v.x);
    m.y = fmaxf(m.y, v.y);
    m.z = fmaxf(m.z, v.z);
    m.w = fmaxf(m.w, v.w);
  }
  ((float4*)pooled)[((size_t)b * NPTS + pt) * (CIN / 4) + lane] = m;
}

// ---------------------------------------------------------------------------
// Kernel 3: fused MLP  out = W2 * relu(W1 * pooled + b1) + b2
// One wave owns NT=2 sixteen-point tiles (32 points): every weight A-fragment
// load now feeds 2 WMMAs (loop mix 10 loads : 16 wmma instead of 9 : 8),
// halving W1/W2 cache traffic.  2 waves / block, 32 KB LDS h-staging.
// WMMA: V_WMMA_F32_16X16X4_F32 (exact fp32 path).
//   A 16x4: lanes 0-15 -> M=lane, v0=K0,v1=K1; lanes 16-31 -> M=lane-16, K2/K3
//   B  4x16: col = lane&15, v0/v1 = rows (half*2, half*2+1)
//   D 16x16: VGPR r -> M = r + (lane<16 ? 0 : 8), N = lane&15
// ---------------------------------------------------------------------------
#define WAVES_PER_BLOCK 2
#define NT 2   // 16-point tiles per wave

__global__ __launch_bounds__(32 * WAVES_PER_BLOCK) void mlp_kernel(
    const float* __restrict__ pooled,
    const float* __restrict__ W1, const float* __restrict__ b1,
    const float* __restrict__ W2, const float* __restrict__ b2,
    float* __restrict__ out) {
  __shared__ float h_lds[WAVES_PER_BLOCK][NT][CIN][16];   // 32 KB

  const int wave = threadIdx.x >> 5;
  const int lane = threadIdx.x & 31;
  const int half = lane >> 4;          // 0: lanes 0-15, 1: lanes 16-31
  const int l16  = lane & 15;

  const int tile0 = (blockIdx.x * WAVES_PER_BLOCK + wave) * NT;
  const int b     = tile0 / (NPTS / 16);      // NPTS/16 = 256 divisible by NT
  const int n0    = (tile0 % (NPTS / 16)) * 16;

  const float* P = pooled + ((size_t)b * NPTS + n0) * CIN;  // 32 rows x 128

  // ---------------- GEMM1: h(128 x 32) = W1(128x128) * P^T(128 x 32) -------
  v8f acc[NT][8];
  #pragma unroll
  for (int t = 0; t < NT; ++t)
    #pragma unroll
    for (int ot = 0; ot < 8; ++ot) acc[t][ot] = (v8f){};

  #pragma unroll 4
  for (int k = 0; k < CIN / 4; ++k) {
    const int c = k * 4 + half * 2;
    v2f bf[NT];
    #pragma unroll
    for (int t = 0; t < NT; ++t) {
      bf[t].x = P[(size_t)(t * 16 + l16) * CIN + c];
      bf[t].y = P[(size_t)(t * 16 + l16) * CIN + c + 1];
    }
    #pragma unroll
    for (int ot = 0; ot < 8; ++ot) {
      const float* wrow = W1 + (size_t)(ot * 16 + l16) * CIN + c;
      v2f af;
      af.x = wrow[0];
      af.y = wrow[1];
      #pragma unroll
      for (int t = 0; t < NT; ++t)
        acc[t][ot] = __builtin_amdgcn_wmma_f32_16x16x4_f32(
            false, af, false, bf[t], (short)0, acc[t][ot], false, false);
    }
  }

  // Fused bias + ReLU, stage h into LDS in (tile, c, n) order.
  #pragma unroll
  for (int t = 0; t < NT; ++t) {
    float* hl = &h_lds[wave][t][0][0];
    #pragma unroll
    for (int ot = 0; ot < 8; ++ot) {
      #pragma unroll
      for (int r = 0; r < 8; ++r) {
        const int c = ot * 16 + r + half * 8;
        hl[c * 16 + l16] = fmaxf(acc[t][ot][r] + b1[c], 0.0f);
      }
    }
  }
  __syncthreads();

  // ---------------- GEMM2: out(256 x 32) = W2(256x128) * h(128 x 32) -------
  float* ob = out + (size_t)b * COUT * NPTS;
  #pragma unroll
  for (int huf = 0; huf < 2; ++huf) {          // two 128-row halves of W2
    v8f acc2[NT][8];
    #pragma unroll
    for (int t = 0; t < NT; ++t)
      #pragma unroll
      for (int ot = 0; ot < 8; ++ot) acc2[t][ot] = (v8f){};

    #pragma unroll 4
    for (int k = 0; k < CIN / 4; ++k) {
      const int c = k * 4 + half * 2;
      v2f bf[NT];
      #pragma unroll
      for (int t = 0; t < NT; ++t) {
        bf[t].x = h_lds[wave][t][c][l16];
        bf[t].y = h_lds[wave][t][c + 1][l16];
      }
      #pragma unroll
      for (int ot = 0; ot < 8; ++ot) {
        const int o = huf * 128 + ot * 16 + l16;
        const float* wrow = W2 + (size_t)o * CIN + c;
        v2f af;
        af.x = wrow[0];
        af.y = wrow[1];
        #pragma unroll
        for (int t = 0; t < NT; ++t)
          acc2[t][ot] = __builtin_amdgcn_wmma_f32_16x16x4_f32(
              false, af, false, bf[t], (short)0, acc2[t][ot], false, false);
      }
    }

    // Fused bias + coalesced store into (B, C_OUT, N).
    #pragma unroll
    for (int t = 0; t < NT; ++t) {
      #pragma unroll
      for (int ot = 0; ot < 8; ++ot) {
        #pragma unroll
        for (int r = 0; r < 8; ++r) {
          const int o = huf * 128 + ot * 16 + r + half * 8;
          ob[(size_t)o * NPTS + n0 + t * 16 + l16] = acc2[t][ot][r] + b2[o];
        }
      }
    }
  }
}

// ---------------------------------------------------------------------------
extern "C" void kernel_launch(void* const* d_in, const int* in_sizes, int n_in,
                              void* d_out, int out_size, void* d_ws, size_t ws_size,
                              hipStream_t stream) {
  const float* x   = (const float*)d_in[0];
  const int*   idx = (const int*)d_in[1];   // JAX default x64-off: int32
  const float* W1  = (const float*)d_in[2];
  const float* b1  = (const float*)d_in[3];
  const float* W2  = (const float*)d_in[4];
  const float* b2  = (const float*)d_in[5];
  float* out = (float*)d_out;

  float* xt     = (float*)d_ws;                          // B*N*C floats
  float* pooled = xt + (size_t)BATCH * NPTS * CIN;       // B*N*C floats

  // 1) transpose x -> xt (B,N,C)
  transpose_kernel<<<dim3(NPTS / 32, CIN / 32, BATCH), dim3(32, 8), 0, stream>>>(x, xt);

  // 2) gather + max-pool -> pooled (B,N,C), float4 path
  pool_kernel<<<dim3(NPTS / 4, BATCH), 128, 0, stream>>>(xt, idx, pooled);

  // 3) fused WMMA MLP -> out (B,C_OUT,N)
  const int waves = BATCH * NPTS / 16 / NT;              // 1024 waves, exact
  mlp_kernel<<<waves / WAVES_PER_BLOCK, 32 * WAVES_PER_BLOCK, 0, stream>>>(
      pooled, W1, b1, W2, b2, out);
}